// ClassificationNet_15582141350385
// MI455X (gfx1250) — compile-verified
//
#include <hip/hip_runtime.h>
#include <math.h>

// ---------------- problem constants (match reference) ----------------
constexpr int B_   = 64;
constexpr int N_   = 1024;
constexpr int NN_  = B_ * N_;        // 65536
constexpr int DEG_ = 16;
constexpr int E_   = NN_ * DEG_;     // 1048576
constexpr int H_   = 128;            // IN == H == 128
constexpr int OUT_ = 10;
constexpr int K1_  = 820;            // ceil(0.8*1024)
constexpr int K2_  = 656;            // ceil(0.8*820)

typedef float v2f __attribute__((ext_vector_type(2)));
typedef float v8f __attribute__((ext_vector_type(8)));

// ---------------------------------------------------------------------
// GEMM via V_WMMA_F32_16X16X4_F32:  D[M x 128] = A[M x 128] * W[128 x 128]
// One wave computes a 16-row x 128-col strip (8 C tiles of 16x16).
// fp32 A 16x4 layout: lane l holds row m=l%16, k = 4*kk + 2*(l/16) + v
// fp32 B 4x16 layout: lane l holds col n=l%16, k = 4*kk + 2*(l/16) + v
// fp32 C 16x16 layout: VGPR r, lane l -> m = r + 8*(l/16), n = l%16
// ---------------------------------------------------------------------
__global__ __launch_bounds__(256) void gemm128_wmma(
    const float* __restrict__ A, const float* __restrict__ W,
    float* __restrict__ D, int M) {
  int wave = (int)((blockIdx.x * blockDim.x + threadIdx.x) >> 5);
  int lane = threadIdx.x & 31;
  int mbase = wave * 16;
  if (mbase >= M) return;
  int lmod = lane & 15;
  int lhi  = lane >> 4;                         // 0 or 1

  v8f c[8];
#pragma unroll
  for (int t = 0; t < 8; ++t) c[t] = v8f{};

  const v2f* arow = (const v2f*)(A + (size_t)(mbase + lmod) * H_);

  for (int kk = 0; kk < 32; ++kk) {
    v2f a = arow[lhi + 2 * kk];                 // k = 4kk+2lhi .. +1
    int k0 = 4 * kk + 2 * lhi;
    const float* wc = W + lmod;
#pragma unroll
    for (int t = 0; t < 8; ++t) {
      v2f b;
      b.x = wc[(size_t)k0 * H_ + t * 16];
      b.y = wc[(size_t)(k0 + 1) * H_ + t * 16];
      c[t] = __builtin_amdgcn_wmma_f32_16x16x4_f32(
          false, a, false, b, (short)0, c[t], false, false);
    }
  }
#pragma unroll
  for (int t = 0; t < 8; ++t)
#pragma unroll
    for (int r = 0; r < 8; ++r)
      D[(size_t)(mbase + r + 8 * lhi) * H_ + t * 16 + lmod] = c[t][r];
}

// ---------------- elementwise / scatter helpers ----------------------
__global__ void zero_kernel(float* p, size_t n) {
  size_t i = (size_t)blockIdx.x * blockDim.x + threadIdx.x;
  size_t st = (size_t)gridDim.x * blockDim.x;
  for (; i < n; i += st) p[i] = 0.f;
}

__global__ void deg_init(float* deg, const float* nmask) {
  int i = blockIdx.x * blockDim.x + threadIdx.x;
  if (i < NN_) deg[i] = nmask ? nmask[i] : 1.f;
}

__global__ void deg_accum(float* deg, const int* __restrict__ src,
                          const int* __restrict__ dst, const float* m) {
  int e = blockIdx.x * blockDim.x + threadIdx.x;
  if (e >= E_) return;
  int d = dst[e];
  float ev = m ? m[src[e]] * m[d] : 1.f;
  if (ev != 0.f) atomicAdd(&deg[d], ev);
}

__global__ void finalize_dinv(float* deg) {  // in-place deg -> rsqrt(deg)
  int i = blockIdx.x * blockDim.x + threadIdx.x;
  if (i >= NN_) return;
  float d = deg[i];
  deg[i] = (d > 0.f) ? rsqrtf(d) : 0.f;
}

__global__ void norm_kernel(float* __restrict__ norm, const float* __restrict__ dinv,
                            const int* __restrict__ src, const int* __restrict__ dst,
                            const float* m) {
  int e = blockIdx.x * blockDim.x + threadIdx.x;
  if (e >= E_) return;
  int s = src[e], d = dst[e];
  float ev = m ? m[s] * m[d] : 1.f;
  norm[e] = dinv[s] * dinv[d] * ev;
}

// out[dst] += hw[src] * norm  over all (edge, feature) pairs
__global__ void scatter_feat(float* __restrict__ out, const float* __restrict__ hw,
                             const int* __restrict__ src, const int* __restrict__ dst,
                             const float* __restrict__ norm) {
  long long idx = (long long)blockIdx.x * blockDim.x + threadIdx.x;
  if (idx >= (long long)E_ * H_) return;
  int e = (int)(idx >> 7);
  int f = (int)(idx & (H_ - 1));
  float nv = norm[e];
  if (nv != 0.f)
    atomicAdd(&out[(size_t)dst[e] * H_ + f], hw[(size_t)src[e] * H_ + f] * nv);
}

// out = relu( (out + hw*dinv^2*nm + b) * nm )
__global__ void gcn_finalize(float* __restrict__ out, const float* __restrict__ hw,
                             const float* __restrict__ dinv, const float* __restrict__ b,
                             const float* nm) {
  long long idx = (long long)blockIdx.x * blockDim.x + threadIdx.x;
  if (idx >= (long long)NN_ * H_) return;
  int i = (int)(idx >> 7);
  int f = (int)(idx & (H_ - 1));
  float nmv = nm ? nm[i] : 1.f;
  float di = dinv[i];
  float v = out[idx] + hw[idx] * di * di * nmv;
  v = (v + b[f]) * nmv;
  out[idx] = fmaxf(v, 0.f);
}

// r[i] = h[i].Wr ; t[i] = h[i].We
__global__ void node_dots(float* __restrict__ r, float* __restrict__ t,
                          const float* __restrict__ h, const float* __restrict__ Wr,
                          const float* __restrict__ We) {
  int i = blockIdx.x * blockDim.x + threadIdx.x;
  if (i >= NN_) return;
  const float* hr = h + (size_t)i * H_;
  float a = 0.f, c = 0.f;
  for (int k = 0; k < H_; ++k) { float v = hr[k]; a += v * Wr[k]; c += v * We[k]; }
  r[i] = a; t[i] = c;
}

__global__ void edge_scalar(float* __restrict__ aggs, const float* __restrict__ t,
                            const int* __restrict__ src, const int* __restrict__ dst,
                            const float* m) {
  int e = blockIdx.x * blockDim.x + threadIdx.x;
  if (e >= E_) return;
  int s = src[e], d = dst[e];
  float ev = m ? m[s] * m[d] : 1.f;
  if (ev != 0.f) atomicAdd(&aggs[d], t[s] * ev);
}

__global__ void score_fin(float* __restrict__ sarr, const float* __restrict__ r,
                          const float* __restrict__ aggs, const float* __restrict__ bs) {
  int i = blockIdx.x * blockDim.x + threadIdx.x;
  if (i < NN_) sarr[i] = tanhf(r[i] + aggs[i] + bs[0]);
}

// Per-graph exact top-K (stable ties like lax.top_k): bitonic sort of
// (score desc, index asc) on 1024 elements in LDS; one block per graph.
__global__ __launch_bounds__(512) void topk_kernel(const float* __restrict__ s,
                                                   const float* nmask,
                                                   float* __restrict__ mout, int K) {
  __shared__ float key[N_];
  __shared__ int   kid[N_];
  int g = blockIdx.x;
  for (int i = threadIdx.x; i < N_; i += 512) {
    float v = s[g * N_ + i];
    if (nmask && nmask[g * N_ + i] == 0.f) v = -INFINITY;
    key[i] = v; kid[i] = i;
  }
  __syncthreads();
  for (int k = 2; k <= N_; k <<= 1) {
    for (int j = k >> 1; j > 0; j >>= 1) {
      for (int t = threadIdx.x; t < N_; t += 512) {
        int l = t ^ j;
        if (l > t) {
          float ka = key[t], kb = key[l];
          int ia = kid[t], ib = kid[l];
          bool precBA = (kb > ka) || (kb == ka && ib < ia);  // b before a (desc)
          bool precAB = (ka > kb) || (ka == kb && ia < ib);
          bool dirFwd = ((t & k) == 0);
          if (dirFwd ? precBA : precAB) {
            key[t] = kb; key[l] = ka; kid[t] = ib; kid[l] = ia;
          }
        }
      }
      __syncthreads();
    }
  }
  for (int rk = threadIdx.x; rk < N_; rk += 512)
    mout[g * N_ + kid[rk]] = (rk < K) ? 1.f : 0.f;
}

__global__ void scale_kernel(float* __restrict__ h, const float* __restrict__ s,
                             const float* __restrict__ m) {
  long long idx = (long long)blockIdx.x * blockDim.x + threadIdx.x;
  if (idx >= (long long)NN_ * H_) return;
  int i = (int)(idx >> 7);
  h[idx] *= s[i] * m[i];
}

// [global_max_pool, global_mean_pool] over kept nodes; block per graph.
__global__ __launch_bounds__(128) void readout_kernel(const float* __restrict__ h,
                                                      const float* __restrict__ m,
                                                      float* __restrict__ xout,
                                                      float kinv) {
  int g = blockIdx.x, f = threadIdx.x;
  const float* hg = h + (size_t)g * N_ * H_;
  const float* mg = m + g * N_;
  float mx = -INFINITY, sm = 0.f;
  for (int nd = 0; nd < N_; ++nd) {
    float v = hg[(size_t)nd * H_ + f];
    float mk = mg[nd];
    if (mk > 0.f) mx = fmaxf(mx, v);
    sm += v * mk;
  }
  xout[g * 2 * H_ + f] = mx;
  xout[g * 2 * H_ + H_ + f] = sm * kinv;
}

// Fused MLP head: z=x1+x2; relu(z@Wfc1+b); @Wfc2+b; log_softmax. Block/graph.
__global__ __launch_bounds__(128) void head_kernel(
    const float* __restrict__ x1, const float* __restrict__ x2,
    const float* __restrict__ Wfc1, const float* __restrict__ bfc1,
    const float* __restrict__ Wfc2, const float* __restrict__ bfc2,
    float* __restrict__ out) {
  __shared__ float z[2 * H_];
  __shared__ float zz[H_];
  __shared__ float lg[OUT_];
  __shared__ float lse;
  int g = blockIdx.x, t = threadIdx.x;
  z[t]        = x1[g * 2 * H_ + t]        + x2[g * 2 * H_ + t];
  z[t + H_]   = x1[g * 2 * H_ + H_ + t]   + x2[g * 2 * H_ + H_ + t];
  __syncthreads();
  float acc = bfc1[t];
  for (int k = 0; k < 2 * H_; ++k) acc += z[k] * Wfc1[k * H_ + t];
  zz[t] = fmaxf(acc, 0.f);
  __syncthreads();
  if (t < OUT_) {
    float a = bfc2[t];
    for (int k = 0; k < H_; ++k) a += zz[k] * Wfc2[k * OUT_ + t];
    lg[t] = a;
  }
  __syncthreads();
  if (t == 0) {
    float mx = lg[0];
    for (int i = 1; i < OUT_; ++i) mx = fmaxf(mx, lg[i]);
    float se = 0.f;
    for (int i = 0; i < OUT_; ++i) se += expf(lg[i] - mx);
    lse = mx + logf(se);
  }
  __syncthreads();
  if (t < OUT_) out[g * OUT_ + t] = lg[t] - lse;
}

// ---------------------------------------------------------------------
extern "C" void kernel_launch(void* const* d_in, const int* in_sizes, int n_in,
                              void* d_out, int out_size, void* d_ws, size_t ws_size,
                              hipStream_t stream) {
  const float* x    = (const float*)d_in[0];
  const int*   ei   = (const int*)d_in[1];
  const int*   src  = ei;
  const int*   dst  = ei + E_;
  const float* W1   = (const float*)d_in[3];
  const float* b1   = (const float*)d_in[4];
  const float* W2   = (const float*)d_in[5];
  const float* b2   = (const float*)d_in[6];
  const float* Wr1  = (const float*)d_in[7];
  const float* We1  = (const float*)d_in[8];
  const float* bs1  = (const float*)d_in[9];
  const float* Wr2  = (const float*)d_in[10];
  const float* We2  = (const float*)d_in[11];
  const float* bs2  = (const float*)d_in[12];
  const float* Wfc1 = (const float*)d_in[13];
  const float* bfc1 = (const float*)d_in[14];
  const float* Wfc2 = (const float*)d_in[15];
  const float* bfc2 = (const float*)d_in[16];
  float* outp = (float*)d_out;

  // workspace layout (floats): 2 big NN*H buffers + edge norm + node scratch
  float* ws   = (float*)d_ws;
  float* hw   = ws;                               // NN*H  (hw1, then hw2)
  float* hbuf = hw + (size_t)NN_ * H_;            // NN*H  (h1,  then h2)
  float* norm = hbuf + (size_t)NN_ * H_;          // E
  float* dinv = norm + E_;                        // NN
  float* sarr = dinv + NN_;                       // NN
  float* m1   = sarr + NN_;                       // NN
  float* m2   = m1 + NN_;                         // NN
  float* rr   = m2 + NN_;                         // NN
  float* tt   = rr + NN_;                         // NN
  float* aggs = tt + NN_;                         // NN
  float* x1   = aggs + NN_;                       // B*2H
  float* x2   = x1 + B_ * 2 * H_;                 // B*2H

  const int TB = 256;
  const int gNN   = NN_ / TB;
  const int gE    = E_ / TB;
  const int gNNH  = (NN_ * H_) / TB;              // 32768
  const long long EH = (long long)E_ * H_;
  const int gEH   = (int)(EH / TB);               // 524288
  const int gGemm = (NN_ / 16) / (TB / 32);       // 512 blocks

  // ---------------- stage 1 ----------------
  gemm128_wmma<<<gGemm, TB, 0, stream>>>(x, W1, hw, NN_);
  zero_kernel<<<2048, TB, 0, stream>>>(hbuf, (size_t)NN_ * H_);
  deg_init<<<gNN, TB, 0, stream>>>(dinv, nullptr);
  deg_accum<<<gE, TB, 0, stream>>>(dinv, src, dst, nullptr);
  finalize_dinv<<<gNN, TB, 0, stream>>>(dinv);
  norm_kernel<<<gE, TB, 0, stream>>>(norm, dinv, src, dst, nullptr);
  scatter_feat<<<gEH, TB, 0, stream>>>(hbuf, hw, src, dst, norm);
  gcn_finalize<<<gNNH, TB, 0, stream>>>(hbuf, hw, dinv, b1, nullptr);

  node_dots<<<gNN, TB, 0, stream>>>(rr, tt, hbuf, Wr1, We1);
  zero_kernel<<<256, TB, 0, stream>>>(aggs, (size_t)NN_);
  edge_scalar<<<gE, TB, 0, stream>>>(aggs, tt, src, dst, nullptr);
  score_fin<<<gNN, TB, 0, stream>>>(sarr, rr, aggs, bs1);
  topk_kernel<<<B_, 512, 0, stream>>>(sarr, nullptr, m1, K1_);
  scale_kernel<<<gNNH, TB, 0, stream>>>(hbuf, sarr, m1);
  readout_kernel<<<B_, H_, 0, stream>>>(hbuf, m1, x1, 1.f / (float)K1_);

  // ---------------- stage 2 ----------------
  gemm128_wmma<<<gGemm, TB, 0, stream>>>(hbuf, W2, hw, NN_);   // hw2 = h1@W2
  zero_kernel<<<2048, TB, 0, stream>>>(hbuf, (size_t)NN_ * H_);
  deg_init<<<gNN, TB, 0, stream>>>(dinv, m1);
  deg_accum<<<gE, TB, 0, stream>>>(dinv, src, dst, m1);
  finalize_dinv<<<gNN, TB, 0, stream>>>(dinv);
  norm_kernel<<<gE, TB, 0, stream>>>(norm, dinv, src, dst, m1);
  scatter_feat<<<gEH, TB, 0, stream>>>(hbuf, hw, src, dst, norm);
  gcn_finalize<<<gNNH, TB, 0, stream>>>(hbuf, hw, dinv, b2, m1);

  node_dots<<<gNN, TB, 0, stream>>>(rr, tt, hbuf, Wr2, We2);
  zero_kernel<<<256, TB, 0, stream>>>(aggs, (size_t)NN_);
  edge_scalar<<<gE, TB, 0, stream>>>(aggs, tt, src, dst, m1);
  score_fin<<<gNN, TB, 0, stream>>>(sarr, rr, aggs, bs2);
  topk_kernel<<<B_, 512, 0, stream>>>(sarr, m1, m2, K2_);
  scale_kernel<<<gNNH, TB, 0, stream>>>(hbuf, sarr, m2);
  readout_kernel<<<B_, H_, 0, stream>>>(hbuf, m2, x2, 1.f / (float)K2_);

  // ---------------- head ----------------
  head_kernel<<<B_, H_, 0, stream>>>(x1, x2, Wfc1, bfc1, Wfc2, bfc2, outp);
}